// ImprovedGNNEncoder_59837484368530
// MI455X (gfx1250) — compile-verified
//
#include <hip/hip_runtime.h>
#include <hip/hip_bf16.h>

// ---------------------------------------------------------------------------
// 3-layer GraphSAGE encoder for gfx1250 (MI455X).
//   layer: h = (mean_agg @ Wl) + (x @ Wr) + b   [fused in one WMMA kernel]
//   l1,l2: relu -> batchnorm (training stats); l2 adds residual x1
// Memory-bound on edge gather/scatter (~5 GB @ 23.3 TB/s); GEMMs are ~10 GFLOP
// total so we use full-precision V_WMMA_F32_16X16X4_F32 (exact fp32 numerics).
// Weights are repacked once per launch into (k,k+1)-interleaved pairs so every
// B fragment is a single contiguous global_load_b64.
// ---------------------------------------------------------------------------

typedef float v2f __attribute__((ext_vector_type(2)));
typedef float v8f __attribute__((ext_vector_type(8)));

#define DFEAT 128
#define EPS 1e-5f

// ---------------- degree / inv-degree ----------------
__global__ void degree_kernel(const int* __restrict__ dst, float* __restrict__ deg,
                              int nEdges) {
    int e = blockIdx.x * blockDim.x + threadIdx.x;
    if (e < nEdges) atomicAdd(&deg[dst[e]], 1.0f);
}

__global__ void invdeg_kernel(const float* __restrict__ deg, float* __restrict__ inv,
                              int n) {
    int i = blockIdx.x * blockDim.x + threadIdx.x;
    if (i < n) inv[i] = 1.0f / fmaxf(deg[i], 1.0f);
}

// ---------------- weight repack: pair-interleave along k ----------------
// Wp[((k/2)*128 + n)*2 + (k&1)] = W[k*128 + n]  -> B frags become one b64 load.
__global__ void pack_weights_kernel(const float* __restrict__ W,
                                    float* __restrict__ Wp) {
    int gid = blockIdx.x * blockDim.x + threadIdx.x;  // 128*128 threads
    int k = gid >> 7;
    int n = gid & 127;
    Wp[((((k >> 1) << 7) + n) << 1) + (k & 1)] = W[gid];
}

// ---------------- edge scatter-add: agg[dst] += x[src] ----------------
// One wave32 per edge; each lane moves a float4 (whole 512B row coalesced).
// Atomic f32 adds resolve in L2 (agg = 25.6MB fits the 192MB L2).
// Edge index is wave-uniform -> readfirstlane so src/dst loads scalarize.
__global__ void scatter_add_kernel(const float* __restrict__ X,
                                   const int* __restrict__ src,
                                   const int* __restrict__ dst,
                                   float* __restrict__ agg, int nEdges) {
    int gid = blockIdx.x * blockDim.x + threadIdx.x;
    int e = gid >> 5;                       // wave-uniform (blocks wave-aligned)
    if (e >= nEdges) return;                // wave-uniform exit
    e = __builtin_amdgcn_readfirstlane(e);
    const int s = src[e];
    const int d = dst[e];
    int c = (threadIdx.x & 31) * 4;
    const float4 v = *(const float4*)(X + (size_t)s * DFEAT + c);
    float* out = agg + (size_t)d * DFEAT + c;
    atomicAdd(out + 0, v.x);
    atomicAdd(out + 1, v.y);
    atomicAdd(out + 2, v.z);
    atomicAdd(out + 3, v.w);
}

// ---------------- fused SAGE GEMM with V_WMMA_F32_16X16X4_F32 ----------------
// H = (agg * inv_deg) @ Wl + X @ Wr + b, optional ReLU (compile-time).
// 8 waves/block; wave handles one 16-row tile x 64 columns (grid.y picks half).
// A frag (16x4 f32, ISA layout): lane l -> row M=l%16, k = 2*(l/16)+{0,1}.
// B frag mirrored: lane l -> col N=l%16, same k pattern; loaded from packed W.
template <int RELU>
__global__ void sage_gemm_kernel(const float* __restrict__ agg,
                                 const float* __restrict__ X,
                                 const float* __restrict__ invdeg,
                                 const float* __restrict__ WlP,
                                 const float* __restrict__ WrP,
                                 const float* __restrict__ bias,
                                 float* __restrict__ H, int nRows) {
    const int lane = threadIdx.x & 31;
    const int wave = threadIdx.x >> 5;
    const int tile = blockIdx.x * 8 + wave;            // 16-row tile index
    const int nTiles = (nRows + 15) >> 4;
    if (tile >= nTiles) return;                        // wave-uniform exit
    const int r0 = tile * 16;
    const int nbase = blockIdx.y * 64;                 // column half

    const int mA = lane & 15;                          // A-frag row in tile
    const int kHalf = (lane >> 4) * 2;                 // A/B-frag k offset
    const int nB = nbase + (lane & 15);                // B-frag column

    const int rowA = min(r0 + mA, nRows - 1);          // clamp for tail tile
    const float idg = invdeg[rowA];
    const float* aRow = agg + (size_t)rowA * DFEAT;
    const float* xRow = X + (size_t)rowA * DFEAT;
    const v2f* blP = (const v2f*)WlP;                  // packed (k-pair, n) pairs
    const v2f* brP = (const v2f*)WrP;

    // bias folded into accumulator init: D[m][n] starts at b[n]
    v8f acc[4];
#pragma unroll
    for (int t = 0; t < 4; ++t) {
        const float bv = bias[nB + t * 16];
#pragma unroll
        for (int i = 0; i < 8; ++i) acc[t][i] = bv;
    }

    for (int k = 0; k < DFEAT; k += 4) {
        v2f aM = *(const v2f*)(aRow + k + kHalf);
        aM.x *= idg;
        aM.y *= idg;
        v2f aX = *(const v2f*)(xRow + k + kHalf);
        const size_t pbase = (size_t)((k + kHalf) >> 1) * DFEAT;  // pair row
#pragma unroll
        for (int t = 0; t < 4; ++t) {
            const v2f bl = blP[pbase + nB + t * 16];
            const v2f br = brP[pbase + nB + t * 16];
            acc[t] = __builtin_amdgcn_wmma_f32_16x16x4_f32(
                false, aM, false, bl, (short)0, acc[t], false, false);
            acc[t] = __builtin_amdgcn_wmma_f32_16x16x4_f32(
                false, aX, false, br, (short)0, acc[t], false, false);
        }
    }

    // D layout: VGPR i -> lanes 0-15: row i, lanes 16-31: row 8+i; col = lane%16.
    const int rowOff = (lane >> 4) * 8;
    float* hBase = H + (size_t)(r0 + rowOff) * DFEAT;
    if (r0 + 16 <= nRows) {  // full tile: branch-free stores (common path)
#pragma unroll
        for (int t = 0; t < 4; ++t) {
            const int n = nB + t * 16;
#pragma unroll
            for (int i = 0; i < 8; ++i) {
                float v = acc[t][i];
                if (RELU) v = fmaxf(v, 0.0f);
                hBase[(size_t)i * DFEAT + n] = v;
            }
        }
    } else {  // tail tile: guard rows
#pragma unroll
        for (int t = 0; t < 4; ++t) {
            const int n = nB + t * 16;
#pragma unroll
            for (int i = 0; i < 8; ++i) {
                float v = acc[t][i];
                if (RELU) v = fmaxf(v, 0.0f);
                if (r0 + rowOff + i < nRows) hBase[(size_t)i * DFEAT + n] = v;
            }
        }
    }
}

// ---------------- batchnorm column statistics ----------------
// block = 128 threads (one per column), each block covers 256 rows, coalesced.
__global__ void colstats_kernel(const float* __restrict__ H, float* __restrict__ sums,
                                float* __restrict__ sumsq, int nRows) {
    const int c = threadIdx.x;
    int r0 = blockIdx.x * 256;
    int rend = min(r0 + 256, nRows);
    float s = 0.0f, q = 0.0f;
    for (int r = r0; r < rend; ++r) {
        float v = H[(size_t)r * DFEAT + c];
        s += v;
        q += v * v;
    }
    atomicAdd(&sums[c], s);
    atomicAdd(&sumsq[c], q);
}

__global__ void bn_coeff_kernel(const float* __restrict__ sums,
                                const float* __restrict__ sumsq,
                                const float* __restrict__ g,
                                const float* __restrict__ be,
                                float* __restrict__ scale, float* __restrict__ shift,
                                float invN) {
    int c = threadIdx.x;  // 128 threads
    float mu = sums[c] * invN;
    float var = sumsq[c] * invN - mu * mu;  // biased variance (matches ref)
    float sc = g[c] * rsqrtf(var + EPS);
    scale[c] = sc;
    shift[c] = be[c] - mu * sc;
}

template <int RESID>
__global__ void bn_apply_kernel(const float* __restrict__ H,
                                const float* __restrict__ scale,
                                const float* __restrict__ shift,
                                const float* __restrict__ resid,
                                float* __restrict__ out, int nRows) {
    int gid = blockIdx.x * blockDim.x + threadIdx.x;
    int r = gid >> 5;
    if (r >= nRows) return;
    int c = (gid & 31) * 4;
    size_t off = (size_t)r * DFEAT + c;
    float4 h = *(const float4*)(H + off);
    float4 o;
    o.x = h.x * scale[c + 0] + shift[c + 0];
    o.y = h.y * scale[c + 1] + shift[c + 1];
    o.z = h.z * scale[c + 2] + shift[c + 2];
    o.w = h.w * scale[c + 3] + shift[c + 3];
    if (RESID) {
        float4 rr = *(const float4*)(resid + off);
        o.x += rr.x;
        o.y += rr.y;
        o.z += rr.z;
        o.w += rr.w;
    }
    *(float4*)(out + off) = o;
}

// ---------------------------------------------------------------------------
extern "C" void kernel_launch(void* const* d_in, const int* in_sizes, int n_in,
                              void* d_out, int out_size, void* d_ws, size_t ws_size,
                              hipStream_t stream) {
    const float* x = (const float*)d_in[0];
    const int* edge = (const int*)d_in[1];
    const float* Wl1 = (const float*)d_in[2];
    const float* Wr1 = (const float*)d_in[3];
    const float* b1 = (const float*)d_in[4];
    const float* Wl2 = (const float*)d_in[5];
    const float* Wr2 = (const float*)d_in[6];
    const float* b2 = (const float*)d_in[7];
    const float* Wl3 = (const float*)d_in[8];
    const float* Wr3 = (const float*)d_in[9];
    const float* b3 = (const float*)d_in[10];
    const float* g1 = (const float*)d_in[11];
    const float* be1 = (const float*)d_in[12];
    const float* g2 = (const float*)d_in[13];
    const float* be2 = (const float*)d_in[14];

    const int N = in_sizes[0] / DFEAT;
    const int E = in_sizes[1] / 2;
    const int* src = edge;
    const int* dst = edge + E;

    // workspace carve-up (floats)
    const size_t WSZ = (size_t)DFEAT * DFEAT;
    float* p = (float*)d_ws;
    float* deg = p;   p += N;
    float* inv = p;   p += N;
    float* aggb = p;  p += (size_t)N * DFEAT;
    float* buf1 = p;  p += (size_t)N * DFEAT;
    float* buf2 = p;  p += (size_t)N * DFEAT;
    float* csum = p;  p += DFEAT;
    float* csq = p;   p += DFEAT;
    float* scl = p;   p += DFEAT;
    float* shf = p;   p += DFEAT;
    float* wP[6];
    for (int i = 0; i < 6; ++i) { wP[i] = p; p += WSZ; }

    const int nTiles = (N + 15) >> 4;
    const dim3 gGemm((nTiles + 7) / 8, 2);
    const int gScat = (E * 32 + 255) / 256;
    const int gEdge = (E + 255) / 256;
    const int gNode = (N + 255) / 256;
    const int gApply = (N * 32 + 255) / 256;
    const int gStats = (N + 255) / 256;
    const int gPack = (int)(WSZ / 256);
    const float invN = 1.0f / (float)N;

    // pack all six weight matrices into k-pair-interleaved layout
    const float* wIn[6] = {Wl1, Wr1, Wl2, Wr2, Wl3, Wr3};
    for (int i = 0; i < 6; ++i)
        pack_weights_kernel<<<gPack, 256, 0, stream>>>(wIn[i], wP[i]);

    // degrees
    hipMemsetAsync(deg, 0, (size_t)N * sizeof(float), stream);
    degree_kernel<<<gEdge, 256, 0, stream>>>(dst, deg, E);
    invdeg_kernel<<<gNode, 256, 0, stream>>>(deg, inv, N);

    // ---- layer 1: x -> buf1 ----
    hipMemsetAsync(aggb, 0, (size_t)N * DFEAT * sizeof(float), stream);
    scatter_add_kernel<<<gScat, 256, 0, stream>>>(x, src, dst, aggb, E);
    sage_gemm_kernel<1><<<gGemm, 256, 0, stream>>>(aggb, x, inv, wP[0], wP[1], b1,
                                                   buf1, N);
    hipMemsetAsync(csum, 0, 2 * DFEAT * sizeof(float), stream);  // csum+csq adjacent
    colstats_kernel<<<gStats, 128, 0, stream>>>(buf1, csum, csq, N);
    bn_coeff_kernel<<<1, 128, 0, stream>>>(csum, csq, g1, be1, scl, shf, invN);
    bn_apply_kernel<0><<<gApply, 256, 0, stream>>>(buf1, scl, shf, nullptr, buf1, N);

    // ---- layer 2: buf1 -> buf2 (residual + buf1) ----
    hipMemsetAsync(aggb, 0, (size_t)N * DFEAT * sizeof(float), stream);
    scatter_add_kernel<<<gScat, 256, 0, stream>>>(buf1, src, dst, aggb, E);
    sage_gemm_kernel<1><<<gGemm, 256, 0, stream>>>(aggb, buf1, inv, wP[2], wP[3], b2,
                                                   buf2, N);
    hipMemsetAsync(csum, 0, 2 * DFEAT * sizeof(float), stream);
    colstats_kernel<<<gStats, 128, 0, stream>>>(buf2, csum, csq, N);
    bn_coeff_kernel<<<1, 128, 0, stream>>>(csum, csq, g2, be2, scl, shf, invN);
    bn_apply_kernel<1><<<gApply, 256, 0, stream>>>(buf2, scl, shf, buf1, buf2, N);

    // ---- layer 3: buf2 -> d_out ----
    hipMemsetAsync(aggb, 0, (size_t)N * DFEAT * sizeof(float), stream);
    scatter_add_kernel<<<gScat, 256, 0, stream>>>(buf2, src, dst, aggb, E);
    sage_gemm_kernel<0><<<gGemm, 256, 0, stream>>>(aggb, buf2, inv, wP[4], wP[5], b3,
                                                   (float*)d_out, N);
}